// HyperGraphAttentionLayer_87136296501907
// MI455X (gfx1250) — compile-verified
//
#include <hip/hip_runtime.h>

#define U_N 200000
#define V_N 100000
#define E_N 500000
#define D_N 96
#define B_N 4096

typedef __attribute__((ext_vector_type(2))) float v2f;
typedef __attribute__((ext_vector_type(8))) float v8f;

// ---- f32 WMMA helper: D = A(16x4) * B(4x16) + C, full fp32 precision ----
__device__ __forceinline__ v8f wmma4(v2f a, v2f b, v8f c) {
  return __builtin_amdgcn_wmma_f32_16x16x4_f32(false, a, false, b, (short)0, c,
                                               false, false);
}

// Accumulate one 16x16 output tile: A is 16 x K (row stride ldA), W is K x 96.
// Lane mapping (ISA 7.12.2, 32-bit A 16x4): M = lane&15, K = 2*(lane>>4)+vgpr.
// B symmetric: N = lane&15, K = 2*(lane>>4)+vgpr.
__device__ __forceinline__ v8f gemm_tile(const float* __restrict__ A, int ldA,
                                         const float* __restrict__ W, int ldW,
                                         int K, int n0, int lane) {
  v8f acc = {};
  const int m = lane & 15;
  const int kh = (lane >> 4) * 2;
  const float* arow = A + (long long)m * ldA;
  const float* wcol = W + n0 + m;
  for (int k0 = 0; k0 < K; k0 += 4) {
    const int kk = k0 + kh;
    v2f a; a[0] = arow[kk];               a[1] = arow[kk + 1];
    v2f b; b[0] = wcol[(long long)kk * ldW]; b[1] = wcol[(long long)(kk + 1) * ldW];
    acc = wmma4(a, b, acc);
  }
  return acc;
}

// ---- monotone uint encoding for float atomic-max ----
__device__ __forceinline__ unsigned fenc(float x) {
  unsigned u = __float_as_uint(x);
  return (u & 0x80000000u) ? ~u : (u | 0x80000000u);
}
__device__ __forceinline__ float fdec(unsigned e) {
  return __uint_as_float((e & 0x80000000u) ? (e ^ 0x80000000u) : ~e);
}
#define FENC_NEG_INF 0x007FFFFFu  // fenc(-inf)

// ============ generic (rows x 96) @ (96 x 96) GEMM, optional bias/relu ======
__global__ void k_gemm96(const float* __restrict__ A, const float* __restrict__ W,
                         const float* __restrict__ bias, float* __restrict__ out,
                         int relu) {
  const int lane = threadIdx.x & 31, wave = threadIdx.x >> 5;
  const long long row0 = (long long)(blockIdx.x * 2 + wave) * 16;
  const float* Arows = A + row0 * D_N;
  for (int nt = 0; nt < 6; ++nt) {
    v8f acc = gemm_tile(Arows, D_N, W, D_N, D_N, nt * 16, lane);
    const int col = nt * 16 + (lane & 15);
    const float bb = bias ? bias[col] : 0.f;
    for (int r = 0; r < 8; ++r) {
      float val = acc[r] + bb;
      if (relu) val = val > 0.f ? val : 0.f;
      out[(row0 + r + 8 * (lane >> 4)) * D_N + col] = val;
    }
  }
}

// ============ v = relu(concat(emb0[vid1], emb2[vid2]) @ Wv + bv) ============
__global__ void k_v(const int* __restrict__ videos, const float* __restrict__ emb0,
                    const float* __restrict__ emb2, const float* __restrict__ Wv,
                    const float* __restrict__ bv, float* __restrict__ vout) {
  __shared__ float tile[32 * 64];
  const long long base = (long long)blockIdx.x * 32;
  for (int idx = threadIdx.x; idx < 32 * 64; idx += 64) {
    const int row = idx >> 6, col = idx & 63;
    const long long vi = base + row;
    float val;
    if (col < 32) val = emb0[(long long)videos[vi * 3 + 1] * 32 + col];
    else          val = emb2[(long long)videos[vi * 3 + 2] * 32 + (col - 32)];
    tile[idx] = val;
  }
  __syncthreads();
  const int lane = threadIdx.x & 31, wave = threadIdx.x >> 5;
  const float* A = tile + wave * 16 * 64;
  const long long row0 = base + wave * 16;
  for (int nt = 0; nt < 6; ++nt) {
    v8f acc = gemm_tile(A, 64, Wv, D_N, 64, nt * 16, lane);
    const int col = nt * 16 + (lane & 15);
    const float bb = bv[col];
    for (int r = 0; r < 8; ++r) {
      float val = acc[r] + bb;
      val = val > 0.f ? val : 0.f;
      vout[(row0 + r + 8 * (lane >> 4)) * D_N + col] = val;
    }
  }
}

// == comb = relu(edge_cat @ We + be) + v[ve]   (loop-invariant across layers) =
__global__ void k_comb(const int* __restrict__ edges, const float* __restrict__ UE,
                       const float* __restrict__ e0, const float* __restrict__ e1,
                       const float* __restrict__ e2, const float* __restrict__ e3,
                       const float* __restrict__ e4, const float* __restrict__ e5,
                       const float* __restrict__ We, const float* __restrict__ be,
                       const float* __restrict__ vbuf, float* __restrict__ comb) {
  __shared__ float tile[32 * 416];  // 52 KB of the 320 KB WGP LDS
  const float* tabs[10] = {e0, e1, e2, e3, e0, e1, e2, e3, e4, e5};
  const long long base = (long long)blockIdx.x * 32;
  for (int idx = threadIdx.x; idx < 32 * 416; idx += 64) {
    const int row = idx / 416, col = idx % 416;
    const long long ei = base + row;
    float val;
    if (col < 96) {
      val = UE[(long long)edges[ei * 13 + 0] * D_N + col];
    } else {
      const int t = (col - 96) >> 5, off = (col - 96) & 31;
      val = tabs[t][(long long)edges[ei * 13 + 3 + t] * 32 + off];
    }
    tile[idx] = val;
  }
  __syncthreads();
  const int lane = threadIdx.x & 31, wave = threadIdx.x >> 5;
  const float* A = tile + wave * 16 * 416;
  const long long row0 = base + wave * 16;
  for (int nt = 0; nt < 6; ++nt) {
    v8f acc = gemm_tile(A, 416, We, D_N, 416, nt * 16, lane);
    const int col = nt * 16 + (lane & 15);
    const float bb = be[col];
    for (int r = 0; r < 8; ++r) {
      const long long er = row0 + r + 8 * (lane >> 4);
      float val = acc[r] + bb;
      val = val > 0.f ? val : 0.f;
      val += vbuf[(long long)edges[er * 13 + 1] * D_N + col];
      comb[er * D_N + col] = val;
    }
  }
}

// ===== k = comb@Wk, val = comb@Wval (shared A fragments, dual accumulators) ==
__global__ void k_kv(const float* __restrict__ comb, const float* __restrict__ Wk,
                     const float* __restrict__ Wv, float* __restrict__ kbuf,
                     float* __restrict__ valbuf) {
  const int lane = threadIdx.x & 31, wave = threadIdx.x >> 5;
  const long long row0 = (long long)(blockIdx.x * 2 + wave) * 16;
  const int m = lane & 15, kh = (lane >> 4) * 2;
  const float* arow = comb + (row0 + m) * D_N;
  for (int nt = 0; nt < 6; ++nt) {
    const int n0 = nt * 16;
    v8f ak = {}, av = {};
    for (int k0 = 0; k0 < D_N; k0 += 4) {
      const int kk = k0 + kh;
      v2f a;  a[0]  = arow[kk];                 a[1]  = arow[kk + 1];
      v2f bk; bk[0] = Wk[kk * D_N + n0 + m];    bk[1] = Wk[(kk + 1) * D_N + n0 + m];
      v2f bv; bv[0] = Wv[kk * D_N + n0 + m];    bv[1] = Wv[(kk + 1) * D_N + n0 + m];
      ak = wmma4(a, bk, ak);
      av = wmma4(a, bv, av);
    }
    const int col = n0 + m;
    for (int r = 0; r < 8; ++r) {
      const long long er = row0 + r + 8 * (lane >> 4);
      kbuf[er * D_N + col]   = ak[r];
      valbuf[er * D_N + col] = av[r];
    }
  }
}

// ===== score = <q[ue],k> / sqrt(32) per head; segment max via atomicMax ======
__global__ void k_score(const int* __restrict__ edges, const float* __restrict__ q,
                        const float* __restrict__ kbuf, float* __restrict__ sc,
                        unsigned* __restrict__ mmax) {
  const long long gid = (long long)blockIdx.x * blockDim.x + threadIdx.x;
  if (gid >= (long long)E_N * 3) return;
  const long long e = gid / 3;
  const int h = (int)(gid % 3);
  const long long ue = edges[e * 13 + 0];
  const float* qp = q + ue * D_N + h * 32;
  const float* kp = kbuf + e * D_N + h * 32;
  float s = 0.f;
#pragma unroll
  for (int d = 0; d < 32; ++d) s += qp[d] * kp[d];
  s *= 0.17677669529663687f;  // 1/sqrt(32)
  sc[gid] = s;
  atomicMax(mmax + ue * 3 + h, fenc(s));
}

// ===== ex = exp(score - m[ue]); segment sum via atomicAdd =====
__global__ void k_exp(const int* __restrict__ edges, const unsigned* __restrict__ mmax,
                      float* __restrict__ sc, float* __restrict__ den) {
  const long long gid = (long long)blockIdx.x * blockDim.x + threadIdx.x;
  if (gid >= (long long)E_N * 3) return;
  const long long e = gid / 3;
  const int h = (int)(gid % 3);
  const long long ue = edges[e * 13 + 0];
  const float ex = expf(sc[gid] - fdec(mmax[ue * 3 + h]));
  sc[gid] = ex;
  atomicAdd(den + ue * 3 + h, ex);
}

// ===== u += segment_sum(alpha * val) via atomicAdd =====
__global__ void k_agg(const int* __restrict__ edges, const float* __restrict__ sc,
                      const float* __restrict__ den, const float* __restrict__ valbuf,
                      float* __restrict__ u) {
  const long long gid = (long long)blockIdx.x * blockDim.x + threadIdx.x;
  if (gid >= (long long)E_N * D_N) return;
  const long long e = gid / D_N;
  const int d = (int)(gid % D_N);
  const int h = d >> 5;
  const long long ue = edges[e * 13 + 0];
  const float alpha = sc[e * 3 + h] / (den[ue * 3 + h] + 1e-9f);
  atomicAdd(u + ue * D_N + d, alpha * valbuf[gid]);
}

// ===== out[b] = (u[idx]*gamma/sqrt(1.001)+beta) @ Wout + bout (B rows only) ==
__global__ void k_out(const int* __restrict__ inputs_idx, const float* __restrict__ u,
                      const float* __restrict__ gamma, const float* __restrict__ beta,
                      const float* __restrict__ Wout, const float* __restrict__ bout,
                      float* __restrict__ out) {
  const int b = blockIdx.x, j = threadIdx.x;  // 64 threads
  const long long row = inputs_idx[b * 2 + 0];
  const float gs = 0.99950037468777324f;  // 1/sqrt(1.001)
  float acc = bout[j];
  for (int d = 0; d < D_N; ++d) {
    const float un = u[row * D_N + d] * (gamma[d] * gs) + beta[d];
    acc += un * Wout[d * 64 + j];
  }
  out[(long long)b * 64 + j] = acc;
}

__global__ void k_fill(unsigned* __restrict__ p, unsigned v, long long n) {
  const long long i = (long long)blockIdx.x * blockDim.x + threadIdx.x;
  if (i < n) p[i] = v;
}

extern "C" void kernel_launch(void* const* d_in, const int* in_sizes, int n_in,
                              void* d_out, int out_size, void* d_ws, size_t ws_size,
                              hipStream_t stream) {
  (void)in_sizes; (void)n_in; (void)out_size; (void)ws_size;
  const float* UE   = (const float*)d_in[0];
  const float* e0   = (const float*)d_in[1];
  const float* e1   = (const float*)d_in[2];
  const float* e2   = (const float*)d_in[3];
  const float* e3   = (const float*)d_in[4];
  const float* e4   = (const float*)d_in[5];
  const float* e5   = (const float*)d_in[6];
  const float* Wu   = (const float*)d_in[7];
  const float* bu   = (const float*)d_in[8];
  const float* Wv   = (const float*)d_in[9];
  const float* bv   = (const float*)d_in[10];
  const float* We   = (const float*)d_in[11];
  const float* be   = (const float*)d_in[12];
  const float* Wq   = (const float*)d_in[13];
  const float* Wk   = (const float*)d_in[14];
  const float* Wval = (const float*)d_in[15];
  const float* gam  = (const float*)d_in[16];
  const float* bet  = (const float*)d_in[17];
  const float* Wout = (const float*)d_in[18];
  const float* bout = (const float*)d_in[19];
  const int* videos = (const int*)d_in[20];
  const int* edges  = (const int*)d_in[21];
  const int* iidx   = (const int*)d_in[22];

  // workspace layout (floats)
  float* ws = (float*)d_ws;
  size_t o = 0;
  float* u    = ws + o; o += (size_t)U_N * D_N;
  float* q    = ws + o; o += (size_t)U_N * D_N;
  float* vb   = ws + o; o += (size_t)V_N * D_N;
  float* comb = ws + o; o += (size_t)E_N * D_N;
  float* kb   = ws + o; o += (size_t)E_N * D_N;
  float* valb = ws + o; o += (size_t)E_N * D_N;
  float* sc   = ws + o; o += (size_t)E_N * 3;
  unsigned* mm = (unsigned*)(ws + o); o += (size_t)U_N * 3;
  float* den  = ws + o; o += (size_t)U_N * 3;

  // feature GEMMs (all fp32 WMMA)
  k_gemm96<<<U_N / 32, 64, 0, stream>>>(UE, Wu, bu, u, 1);
  k_v<<<V_N / 32, 64, 0, stream>>>(videos, e0, e2, Wv, bv, vb);
  k_comb<<<E_N / 32, 64, 0, stream>>>(edges, UE, e0, e1, e2, e3, e4, e5, We, be, vb, comb);

  const long long eh = (long long)E_N * 3;
  const long long ed = (long long)E_N * D_N;
  const long long uh = (long long)U_N * 3;
  for (int l = 0; l < 3; ++l) {
    const size_t wo = (size_t)l * D_N * D_N;
    k_gemm96<<<U_N / 32, 64, 0, stream>>>(u, Wq + wo, nullptr, q, 0);
    k_kv<<<E_N / 32, 64, 0, stream>>>(comb, Wk + wo, Wval + wo, kb, valb);
    k_fill<<<(unsigned)((uh + 255) / 256), 256, 0, stream>>>(mm, FENC_NEG_INF, uh);
    k_fill<<<(unsigned)((uh + 255) / 256), 256, 0, stream>>>((unsigned*)den, 0u, uh);
    k_score<<<(unsigned)((eh + 255) / 256), 256, 0, stream>>>(edges, q, kb, sc, mm);
    k_exp<<<(unsigned)((eh + 255) / 256), 256, 0, stream>>>(edges, mm, sc, den);
    k_agg<<<(unsigned)((ed + 255) / 256), 256, 0, stream>>>(edges, sc, den, valb, u);
  }

  k_out<<<B_N, 64, 0, stream>>>(iidx, u, gam, bet, Wout, bout, (float*)d_out);
}